// KGReasoning_7962869367574
// MI455X (gfx1250) — compile-verified
//
#include <hip/hip_runtime.h>
#include <stdint.h>

// ---------------------------------------------------------------------------
// KGReasoning relation projection: column-wise max / first-occurrence argmax of
// premax[i,j] = e[i] * R[i,j]  (R optionally mapped to 1 - min(1, R)).
//
// Roofline: R = 256 MB read once -> ~11 us at 23.3 TB/s. Pure bandwidth.
// Max-plus semiring => WMMA not applicable; we use the CDNA5 async
// global->LDS pipeline (ASYNCcnt), triple-buffered so each wave keeps two
// 16 KB chunks (8 async b128 ops) in flight while computing a third.
// ---------------------------------------------------------------------------

#define N                8192
#define TPB              256
#define VPT              4                       // columns per thread (float4)
#define COLS_PER_BLOCK   (TPB * VPT)             // 1024
#define NCOLBLK          (N / COLS_PER_BLOCK)    // 8
#define ROWS_PER_STRIPE  128
#define NSTRIPES         (N / ROWS_PER_STRIPE)   // 64
#define CHUNK            4                       // rows per async chunk
#define NCHUNK           (ROWS_PER_STRIPE / CHUNK)
#define NBUF             3                       // async pipeline depth

#if defined(__HIP_DEVICE_COMPILE__) && __has_builtin(__builtin_amdgcn_global_load_async_to_lds_b128)
#define USE_ASYNC 1
#else
#define USE_ASYNC 0
#endif

// The async-to-LDS builtin takes (v4i AS1*, v4i AS3*, imm offset, imm cpol).
typedef int v4i_t __attribute__((ext_vector_type(4)));
typedef __attribute__((address_space(1))) v4i_t gv4i_t;   // global b128
typedef __attribute__((address_space(3))) v4i_t lv4i_t;   // LDS b128

// Packed merge key: higher value wins; ties -> smaller row index wins
// (matches running max with strict '>' over ascending i). All premax values
// are >= 0, so raw float bits are monotone under unsigned compare.
__device__ __forceinline__ unsigned long long pack_key(float v, int i) {
    return ((unsigned long long)__float_as_uint(v) << 32) |
           (unsigned long long)(0xFFFFFFFFu - (unsigned)i);
}

__device__ __forceinline__ void upd(float& mx, int& id, float r, float e,
                                    int i, bool neg) {
    if (neg) r = 1.0f - fminf(1.0f, r);
    float p = e * r;
    if (p > mx) { mx = p; id = i; }
}

template <int CNT>
__device__ __forceinline__ void wait_async() {
#if USE_ASYNC
#if __has_builtin(__builtin_amdgcn_s_wait_asynccnt)
    __builtin_amdgcn_s_wait_asynccnt(CNT);
#else
    asm volatile("s_wait_asynccnt %0" ::"i"(CNT) : "memory");
#endif
#endif
}

// ---------------------------------------------------------------------------
__global__ void kg_init_ws(unsigned long long* __restrict__ ws) {
    int j = blockIdx.x * blockDim.x + threadIdx.x;
    if (j < N) ws[j] = 0ull;   // beaten by every real candidate (i=0 key > 0)
}

// ---------------------------------------------------------------------------
// One (stripe, column-block) tile per workgroup.  blockIdx.x = stripe*NCOLBLK+cb
__global__ __launch_bounds__(TPB) void kg_stripe(
        const float* __restrict__ emb, const float* __restrict__ R,
        const int* __restrict__ is_neg_p,
        unsigned long long* __restrict__ ws) {
    const int colblk = blockIdx.x % NCOLBLK;
    const int stripe = blockIdx.x / NCOLBLK;
    const int gcol0  = colblk * COLS_PER_BLOCK;
    const int col0   = gcol0 + threadIdx.x * VPT;
    const int row0   = stripe * ROWS_PER_STRIPE;
    const bool neg   = (is_neg_p[0] != 0);

    __shared__ float eL[ROWS_PER_STRIPE];
    if (threadIdx.x < ROWS_PER_STRIPE) eL[threadIdx.x] = emb[row0 + threadIdx.x];
    __syncthreads();

    float mx0 = -1.0f, mx1 = -1.0f, mx2 = -1.0f, mx3 = -1.0f;
    int   id0 = row0,  id1 = row0,  id2 = row0,  id3 = row0;

#if USE_ASYNC
    // Triple-buffered async global->LDS staging (CDNA5 ASYNCcnt pipeline).
    __shared__ float buf[NBUF][CHUNK][COLS_PER_BLOCK];   // 3 x 16 KB

    // issue one CHUNK (4 rows x 4 KB) with per-lane b128 async copies
    auto issue = [&](int c, int pb) {
#pragma unroll
        for (int q = 0; q < CHUNK; ++q) {
            const float* gp = R + (size_t)(row0 + c * CHUNK + q) * N
                                + gcol0 + threadIdx.x * VPT;
            float* lp = &buf[pb][q][threadIdx.x * VPT];
            // two-step casts: reinterpret pointee, then address-space cast
            const v4i_t* gpv = reinterpret_cast<const v4i_t*>(gp);
            v4i_t*       lpv = reinterpret_cast<v4i_t*>(lp);
            __builtin_amdgcn_global_load_async_to_lds_b128(
                (gv4i_t*)gpv, (lv4i_t*)lpv, 0, 0);
        }
    };

    issue(0, 0);
    issue(1, 1);
    for (int c = 0; c < NCHUNK; ++c) {
        // Async loads of one wave complete in order: if chunk c+1 is still in
        // flight (4 newest ops), waiting for ASYNCcnt <= CHUNK guarantees
        // chunk c has fully landed.  Drain completely on the last chunk.
        if (c + 1 < NCHUNK) wait_async<CHUNK>(); else wait_async<0>();
        __syncthreads();     // all waves landed chunk c; all done with c-1
        if (c + 2 < NCHUNK) issue(c + 2, (c + 2) % NBUF);
        const int cb = c % NBUF;
#pragma unroll
        for (int q = 0; q < CHUNK; ++q) {
            const int i = row0 + c * CHUNK + q;
            const float4 v =
                *reinterpret_cast<const float4*>(&buf[cb][q][threadIdx.x * VPT]);
            const float e = eL[c * CHUNK + q];
            upd(mx0, id0, v.x, e, i, neg);
            upd(mx1, id1, v.y, e, i, neg);
            upd(mx2, id2, v.z, e, i, neg);
            upd(mx3, id3, v.w, e, i, neg);
        }
    }
#else
    // Direct streaming path: coalesced b128 loads + gfx1250 prefetch.
#pragma unroll 4
    for (int r = 0; r < ROWS_PER_STRIPE; ++r) {
        const int i = row0 + r;
        const float* gp = R + (size_t)i * N + col0;
        if (r + 8 < ROWS_PER_STRIPE)
            __builtin_prefetch(gp + (size_t)8 * N, 0, 0);
        const float4 v = *reinterpret_cast<const float4*>(gp);
        const float e  = eL[r];
        upd(mx0, id0, v.x, e, i, neg);
        upd(mx1, id1, v.y, e, i, neg);
        upd(mx2, id2, v.z, e, i, neg);
        upd(mx3, id3, v.w, e, i, neg);
    }
#endif

    // One packed u64 atomic max per owned column merges stripes globally.
    atomicMax(&ws[col0 + 0], pack_key(mx0, id0));
    atomicMax(&ws[col0 + 1], pack_key(mx1, id1));
    atomicMax(&ws[col0 + 2], pack_key(mx2, id2));
    atomicMax(&ws[col0 + 3], pack_key(mx3, id3));
}

// ---------------------------------------------------------------------------
__global__ void kg_finalize(const unsigned long long* __restrict__ ws,
                            float* __restrict__ outv, int* __restrict__ outi) {
    int j = blockIdx.x * blockDim.x + threadIdx.x;
    if (j >= N) return;
    unsigned long long k = ws[j];
    outv[j] = __uint_as_float((unsigned)(k >> 32));
    outi[j] = (int)(0xFFFFFFFFu - (unsigned)(k & 0xFFFFFFFFull));
}

// ---------------------------------------------------------------------------
// Fallback if the harness workspace is too small for the packed keys:
// each block owns a column block and scans all rows (fewer blocks, no ws).
__global__ __launch_bounds__(TPB) void kg_full_direct(
        const float* __restrict__ emb, const float* __restrict__ R,
        const int* __restrict__ is_neg_p,
        float* __restrict__ outv, int* __restrict__ outi) {
    const int col0 = blockIdx.x * COLS_PER_BLOCK + threadIdx.x * VPT;
    const bool neg = (is_neg_p[0] != 0);

    float mx0 = -1.0f, mx1 = -1.0f, mx2 = -1.0f, mx3 = -1.0f;
    int   id0 = 0,     id1 = 0,     id2 = 0,     id3 = 0;

    __shared__ float eL[TPB];
    for (int t = 0; t < N; t += TPB) {
        __syncthreads();
        eL[threadIdx.x] = emb[t + threadIdx.x];
        __syncthreads();
#pragma unroll 4
        for (int r = 0; r < TPB; ++r) {
            const int i = t + r;
            const float4 v = *reinterpret_cast<const float4*>(
                R + (size_t)i * N + col0);
            const float e = eL[r];
            upd(mx0, id0, v.x, e, i, neg);
            upd(mx1, id1, v.y, e, i, neg);
            upd(mx2, id2, v.z, e, i, neg);
            upd(mx3, id3, v.w, e, i, neg);
        }
    }
    outv[col0 + 0] = mx0; outi[col0 + 0] = id0;
    outv[col0 + 1] = mx1; outi[col0 + 1] = id1;
    outv[col0 + 2] = mx2; outi[col0 + 2] = id2;
    outv[col0 + 3] = mx3; outi[col0 + 3] = id3;
}

// ---------------------------------------------------------------------------
extern "C" void kernel_launch(void* const* d_in, const int* in_sizes, int n_in,
                              void* d_out, int out_size, void* d_ws, size_t ws_size,
                              hipStream_t stream) {
    const float* emb    = (const float*)d_in[0];   // [1, 8192] f32
    const float* R      = (const float*)d_in[1];   // [8192, 8192] f32
    const int*   is_neg = (const int*)d_in[2];     // scalar int

    float* outv = (float*)d_out;                   // [8192] f32 maxima
    int*   outi = (int*)((float*)d_out + N);       // [8192] i32 argmax

    if (ws_size >= (size_t)N * sizeof(unsigned long long)) {
        unsigned long long* ws = (unsigned long long*)d_ws;
        kg_init_ws<<<N / TPB, TPB, 0, stream>>>(ws);
        kg_stripe<<<NSTRIPES * NCOLBLK, TPB, 0, stream>>>(emb, R, is_neg, ws);
        kg_finalize<<<N / TPB, TPB, 0, stream>>>(ws, outv, outi);
    } else {
        kg_full_direct<<<NCOLBLK, TPB, 0, stream>>>(emb, R, is_neg, outv, outi);
    }
}